// SimpleGNN_44633300140823
// MI455X (gfx1250) — compile-verified
//
#include <hip/hip_runtime.h>

#define HID 128
#define NGRAPH 2048

typedef __attribute__((ext_vector_type(2))) float v2f;
typedef __attribute__((ext_vector_type(8))) float v8f;

// ---------------------------------------------------------------- utilities
__global__ void gcn_fill_f32(float* __restrict__ p, float v, long long n) {
  long long i = (long long)blockIdx.x * blockDim.x + threadIdx.x;
  if (i < n) p[i] = v;
}

// deg[dst[e]] += 1 (deg buffer pre-filled with 1.0 for the self loop)
__global__ void gcn_degree(const int* __restrict__ dst, float* __restrict__ deg,
                           long long nE) {
  long long e = (long long)blockIdx.x * blockDim.x + threadIdx.x;
  if (e < nE) unsafeAtomicAdd(&deg[dst[e]], 1.0f);
}

__global__ void gcn_rsqrt(float* __restrict__ deg, int n) {
  int i = blockIdx.x * blockDim.x + threadIdx.x;
  if (i < n) deg[i] = rsqrtf(deg[i]);   // deg >= 1 (self loop), no zero case
}

// norm[e] = dinv[src]*dinv[dst]; entries [nE, nE+N) are the self loops
__global__ void gcn_norm(const int* __restrict__ src, const int* __restrict__ dst,
                         const float* __restrict__ dinv, float* __restrict__ nrm,
                         long long nE, long long ntot) {
  long long e = (long long)blockIdx.x * blockDim.x + threadIdx.x;
  if (e >= ntot) return;
  int s, d;
  if (e < nE) { s = src[e]; d = dst[e]; } else { s = d = (int)(e - nE); }
  nrm[e] = dinv[s] * dinv[d];
}

// ------------------------------------------------- layer 1: fused outer-product scatter
// msg[e][j] = x[src[e]] * W1[j] * norm[e]; acc[dst[e]][j] += msg
// one thread per (edge, 4-feature chunk): idx = e*32 + chunk
__global__ void gcn_scatter_l1(const float* __restrict__ x, const float* __restrict__ W1,
                               const int* __restrict__ src, const int* __restrict__ dst,
                               const float* __restrict__ nrm, float* __restrict__ acc,
                               long long ntot, long long nE) {
  long long idx = (long long)blockIdx.x * blockDim.x + threadIdx.x;
  long long e = idx >> 5;
  int chunk = (int)(idx & 31);
  if (e >= ntot) return;
  int s, d;
  if (e < nE) { s = src[e]; d = dst[e]; } else { s = d = (int)(e - nE); }
  float coef = x[s] * nrm[e];
  float4 w = *(const float4*)(W1 + chunk * 4);
  float* out = acc + (size_t)d * HID + chunk * 4;
  unsafeAtomicAdd(out + 0, coef * w.x);
  unsafeAtomicAdd(out + 1, coef * w.y);
  unsafeAtomicAdd(out + 2, coef * w.z);
  unsafeAtomicAdd(out + 3, coef * w.w);
}

// ------------------------------------------------- layers 2/3: gather-scale-scatter
__global__ void gcn_scatter(const float* __restrict__ hw, const int* __restrict__ src,
                            const int* __restrict__ dst, const float* __restrict__ nrm,
                            float* __restrict__ acc, long long ntot, long long nE) {
  long long idx = (long long)blockIdx.x * blockDim.x + threadIdx.x;
  long long e = idx >> 5;
  int chunk = (int)(idx & 31);
  if (e >= ntot) return;
  int s, d;
  if (e < nE) { s = src[e]; d = dst[e]; } else { s = d = (int)(e - nE); }
  float c = nrm[e];
  float4 m = *(const float4*)(hw + (size_t)s * HID + chunk * 4);
  float* out = acc + (size_t)d * HID + chunk * 4;
  unsafeAtomicAdd(out + 0, c * m.x);
  unsafeAtomicAdd(out + 1, c * m.y);
  unsafeAtomicAdd(out + 2, c * m.z);
  unsafeAtomicAdd(out + 3, c * m.w);
}

__global__ void gcn_bias_relu(const float* __restrict__ in, const float* __restrict__ b,
                              float* __restrict__ out, long long rows) {
  long long i = (long long)blockIdx.x * blockDim.x + threadIdx.x;
  if (i >= rows * HID) return;
  float v = in[i] + b[(int)(i & (HID - 1))];
  out[i] = v > 0.0f ? v : 0.0f;
}

// ------------------------------------------------- FP32 WMMA GEMM: C[rows,128] = A[rows,128] @ W[128,128]
// rows must be a multiple of 16 (100000 and 2048 both are).
// block = 256 threads = 8 waves; each wave owns one 16-row tile, all 8 col tiles.
// W staged in LDS (64 KB) once per block.
__global__ void gcn_gemm128(const float* __restrict__ A, const float* __restrict__ W,
                            float* __restrict__ C, int rows) {
  __shared__ float lW[HID * HID];   // 64 KB
  for (int i = threadIdx.x; i < HID * HID; i += blockDim.x) lW[i] = W[i];
  __syncthreads();

  const int wave = threadIdx.x >> 5;
  const int lane = threadIdx.x & 31;
  const int tile = blockIdx.x * 8 + wave;        // 16-row tile index
  if (tile * 16 >= rows) return;                 // wave-uniform

  const int m = lane & 15;                       // row within tile / col within col-tile
  const int khalf = (lane >> 4) * 2;             // K sub-offset per ISA A/B f32 layout
  const float* arow = A + (size_t)(tile * 16 + m) * HID;

  v8f acc[8];
#pragma unroll
  for (int t = 0; t < 8; ++t) acc[t] = (v8f){};

  for (int k = 0; k < HID; k += 4) {
    v2f a;
    a.x = arow[k + khalf];
    a.y = arow[k + khalf + 1];
#pragma unroll
    for (int t = 0; t < 8; ++t) {
      const int col = t * 16 + m;
      v2f b;
      b.x = lW[(k + khalf) * HID + col];
      b.y = lW[(k + khalf + 1) * HID + col];
      acc[t] = __builtin_amdgcn_wmma_f32_16x16x4_f32(
          /*neg_a=*/false, a, /*neg_b=*/false, b,
          /*c_mod=*/(short)0, acc[t], /*reuse_a=*/false, /*reuse_b=*/false);
    }
  }

#pragma unroll
  for (int t = 0; t < 8; ++t) {
    const int col = t * 16 + m;
#pragma unroll
    for (int v = 0; v < 8; ++v) {
      const int r = tile * 16 + v + 8 * (lane >> 4);
      C[(size_t)r * HID + col] = acc[t][v];
    }
  }
}

// ------------------------------------------------- pooling
__global__ void gcn_pool(const float* __restrict__ h, const int* __restrict__ batch,
                         float* __restrict__ sums, float* __restrict__ cnt, long long n) {
  long long idx = (long long)blockIdx.x * blockDim.x + threadIdx.x;
  long long node = idx >> 5;
  int chunk = (int)(idx & 31);
  if (node >= n) return;
  int g = batch[node];
  float4 v = *(const float4*)(h + (size_t)node * HID + chunk * 4);
  float* out = sums + (size_t)g * HID + chunk * 4;
  unsafeAtomicAdd(out + 0, v.x);
  unsafeAtomicAdd(out + 1, v.y);
  unsafeAtomicAdd(out + 2, v.z);
  unsafeAtomicAdd(out + 3, v.w);
  if (chunk == 0) unsafeAtomicAdd(&cnt[g], 1.0f);
}

__global__ void gcn_mean(const float* __restrict__ sums, const float* __restrict__ cnt,
                         float* __restrict__ g, int ngraph) {
  int i = blockIdx.x * blockDim.x + threadIdx.x;
  if (i >= ngraph * HID) return;
  float c = cnt[i >> 7];
  g[i] = sums[i] / (c > 1.0f ? c : 1.0f);
}

// ------------------------------------------------- head: out[g] = g2[g] . Wl3 + bl3
__global__ void gcn_head(const float* __restrict__ g, const float* __restrict__ Wl3,
                         const float* __restrict__ bl3, float* __restrict__ out, int ngraph) {
  int i = blockIdx.x * blockDim.x + threadIdx.x;
  if (i >= ngraph) return;
  float s = 0.0f;
  const float* gr = g + (size_t)i * HID;
#pragma unroll 8
  for (int j = 0; j < HID; ++j) s += gr[j] * Wl3[j];
  out[i] = s + bl3[0];
}

// ---------------------------------------------------------------- launch
extern "C" void kernel_launch(void* const* d_in, const int* in_sizes, int n_in,
                              void* d_out, int out_size, void* d_ws, size_t ws_size,
                              hipStream_t stream) {
  const float* x    = (const float*)d_in[0];
  const int*   edge = (const int*)d_in[1];
  const int*   bat  = (const int*)d_in[2];
  const float* W1   = (const float*)d_in[3];
  const float* b1   = (const float*)d_in[4];
  const float* W2   = (const float*)d_in[5];
  const float* b2   = (const float*)d_in[6];
  const float* W3   = (const float*)d_in[7];
  const float* b3   = (const float*)d_in[8];
  const float* Wl1  = (const float*)d_in[9];
  const float* bl1  = (const float*)d_in[10];
  const float* Wl2  = (const float*)d_in[11];
  const float* bl2  = (const float*)d_in[12];
  const float* Wl3  = (const float*)d_in[13];
  const float* bl3  = (const float*)d_in[14];
  float* out = (float*)d_out;

  const long long N = in_sizes[0];
  const long long E = in_sizes[1] / 2;
  const int* src = edge;
  const int* dst = edge + E;
  const long long ntot = E + N;
  const int G = out_size;                 // 2048 graphs

  // workspace carve-up (256B aligned regions)
  char* wsp = (char*)d_ws;
  auto carve = [&](size_t bytes) -> float* {
    float* p = (float*)wsp;
    wsp += (bytes + 255) & ~(size_t)255;
    return p;
  };
  float* H    = carve((size_t)N * HID * 4);      // feature / accumulator buffer
  float* HW   = carve((size_t)N * HID * 4);      // h @ W
  float* dinv = carve((size_t)N * 4);            // deg -> dinv in place
  float* nrm  = carve((size_t)ntot * 4);
  float* sums = carve((size_t)G * HID * 4);
  float* cnt  = carve((size_t)G * 4);
  float* gbuf = carve((size_t)G * HID * 4);
  float* gtmp = carve((size_t)G * HID * 4);
  (void)ws_size; (void)n_in;

  const int B = 256;
  auto nb = [](long long work, int b) { return (unsigned)((work + b - 1) / b); };

  // --- normalization coefficients
  gcn_fill_f32<<<nb(N, B), B, 0, stream>>>(dinv, 1.0f, N);              // self loop
  gcn_degree<<<nb(E, B), B, 0, stream>>>(dst, dinv, E);
  gcn_rsqrt<<<nb(N, B), B, 0, stream>>>(dinv, (int)N);
  gcn_norm<<<nb(ntot, B), B, 0, stream>>>(src, dst, dinv, nrm, E, ntot);

  const long long scat = ntot * 32;       // (edge, 4-feature-chunk) threads
  const long long feat = N * HID;
  const unsigned gemmN = (unsigned)(((N / 16) + 7) / 8);
  const unsigned gemmG = (unsigned)(((G / 16) + 7) / 8);

  // --- layer 1 (rank-1 GEMM fused into scatter)
  gcn_fill_f32<<<nb(feat, B), B, 0, stream>>>(H, 0.0f, feat);
  gcn_scatter_l1<<<nb(scat, B), B, 0, stream>>>(x, W1, src, dst, nrm, H, ntot, E);
  gcn_bias_relu<<<nb(feat, B), B, 0, stream>>>(H, b1, H, N);

  // --- layer 2
  gcn_gemm128<<<gemmN, B, 0, stream>>>(H, W2, HW, (int)N);
  gcn_fill_f32<<<nb(feat, B), B, 0, stream>>>(H, 0.0f, feat);
  gcn_scatter<<<nb(scat, B), B, 0, stream>>>(HW, src, dst, nrm, H, ntot, E);
  gcn_bias_relu<<<nb(feat, B), B, 0, stream>>>(H, b2, H, N);

  // --- layer 3
  gcn_gemm128<<<gemmN, B, 0, stream>>>(H, W3, HW, (int)N);
  gcn_fill_f32<<<nb(feat, B), B, 0, stream>>>(H, 0.0f, feat);
  gcn_scatter<<<nb(scat, B), B, 0, stream>>>(HW, src, dst, nrm, H, ntot, E);
  gcn_bias_relu<<<nb(feat, B), B, 0, stream>>>(H, b3, H, N);

  // --- global mean pool
  gcn_fill_f32<<<nb((long long)G * HID, B), B, 0, stream>>>(sums, 0.0f, (long long)G * HID);
  gcn_fill_f32<<<nb(G, B), B, 0, stream>>>(cnt, 0.0f, G);
  gcn_pool<<<nb(N * 32, B), B, 0, stream>>>(H, bat, sums, cnt, N);
  gcn_mean<<<nb((long long)G * HID, B), B, 0, stream>>>(sums, cnt, gbuf, G);

  // --- MLP head (WMMA GEMMs on 2048x128)
  gcn_gemm128<<<gemmG, B, 0, stream>>>(gbuf, Wl1, gtmp, G);
  gcn_bias_relu<<<nb((long long)G * HID, B), B, 0, stream>>>(gtmp, bl1, gbuf, G);
  gcn_gemm128<<<gemmG, B, 0, stream>>>(gbuf, Wl2, gtmp, G);
  gcn_bias_relu<<<nb((long long)G * HID, B), B, 0, stream>>>(gtmp, bl2, gbuf, G);
  gcn_head<<<nb(G, B), B, 0, stream>>>(gbuf, Wl3, bl3, out, G);
}